// GCN_10660108828810
// MI455X (gfx1250) — compile-verified
//
#include <hip/hip_runtime.h>
#include <hip/hip_bf16.h>

typedef __attribute__((ext_vector_type(2))) float v2f;
typedef __attribute__((ext_vector_type(8))) float v8f;

#define HID 64

// ---------------------------------------------------------------------------
// Degree count: dinv[d] += 1 for every edge destination (self-loop added later)
// ---------------------------------------------------------------------------
__global__ void gcn_deg_count(const int* __restrict__ dst, float* __restrict__ dinv, int E) {
    int t = blockIdx.x * blockDim.x + threadIdx.x;
    if (t < E) atomicAdd(&dinv[dst[t]], 1.0f);
}

// dinv[i] = rsqrt(deg_in[i] + 1)   (self-loop guarantees deg > 0)
__global__ void gcn_deg_finish(float* __restrict__ dinv, int n) {
    int i = blockIdx.x * blockDim.x + threadIdx.x;
    if (i < n) dinv[i] = rsqrtf(dinv[i] + 1.0f);
}

// ---------------------------------------------------------------------------
// GEMM: H[n x 64] = X[n x 64] @ W[64 x 64] using V_WMMA_F32_16X16X4_F32.
// 256 threads = 8 waves per block; each wave owns a full 16x64 output
// row-block (four 16x16 tiles, four v8f accumulators). The A fragment is
// loaded once per K-step and reused across all 4 column tiles.
//
// W is staged in LDS K-pair interleaved: Ws2[j*64 + n] = {W[2j][n], W[2j+1][n]}
// so each B fragment (which needs K rows k and k+1 at column n) is a single
// naturally-aligned ds_load_b64 into an even VGPR pair — no packing movs.
// ---------------------------------------------------------------------------
__global__ void __launch_bounds__(256)
gcn_gemm64_wmma(const float* __restrict__ X, const float* __restrict__ W,
                float* __restrict__ H, int n) {
    __shared__ float2 Ws2[(HID / 2) * HID];  // 16 KB of the 320 KB/WGP LDS
    for (int i = threadIdx.x; i < (HID / 2) * HID; i += 256) {
        int j = i >> 6;          // K-pair index: rows 2j, 2j+1
        int col = i & 63;
        float2 p;
        p.x = W[(2 * j)     * HID + col];
        p.y = W[(2 * j + 1) * HID + col];
        Ws2[i] = p;
    }
    __syncthreads();

    const int wave = threadIdx.x >> 5;            // 0..7
    const int lane = threadIdx.x & 31;            // 0..31
    const int row0 = (blockIdx.x * 8 + wave) * 16;
    if (row0 >= n) return;                        // wave-uniform: EXEC stays all-1s

    const int m  = lane & 15;                     // A row within tile (both halves)
    const int kh = (lane >> 4) << 1;              // K sub-offset: 0 (lanes 0-15), 2 (16-31)
    const int nl = lane & 15;                     // B/D column within a 16-col tile

    int mrow = row0 + m; if (mrow >= n) mrow = n - 1;   // clamp (n % 16 == 0 anyway)
    const float* __restrict__ xrow = X + (size_t)mrow * HID;
    // B base for this lane: K-pair (kh/2) at column nl; step per K-iter = 2 pairs = 128 float2 (256 floats)
    const float2* __restrict__ wbase = Ws2 + (size_t)(kh >> 1) * HID + nl;

    v8f acc0 = {}, acc1 = {}, acc2 = {}, acc3 = {};
    #pragma unroll
    for (int k = 0; k < HID; k += 4) {
        // A: 16x4 tile of X (lane l holds row m; K = k+kh, k+kh+1) — loaded ONCE
        v2f a;
        { const float2 av = *(const float2*)(xrow + k + kh); a.x = av.x; a.y = av.y; }
        // B: 4x16 tiles of W, one ds_load_b64 per tile (offsets fold into DS imm)
        const float2* wrow = wbase + (size_t)(k >> 1) * HID;
        v2f b;
        { const float2 bv = wrow[ 0]; b.x = bv.x; b.y = bv.y; }
        acc0 = __builtin_amdgcn_wmma_f32_16x16x4_f32(false, a, false, b, (short)0, acc0, false, false);
        { const float2 bv = wrow[16]; b.x = bv.x; b.y = bv.y; }
        acc1 = __builtin_amdgcn_wmma_f32_16x16x4_f32(false, a, false, b, (short)0, acc1, false, false);
        { const float2 bv = wrow[32]; b.x = bv.x; b.y = bv.y; }
        acc2 = __builtin_amdgcn_wmma_f32_16x16x4_f32(false, a, false, b, (short)0, acc2, false, false);
        { const float2 bv = wrow[48]; b.x = bv.x; b.y = bv.y; }
        acc3 = __builtin_amdgcn_wmma_f32_16x16x4_f32(false, a, false, b, (short)0, acc3, false, false);
    }

    // D layout: VGPR r -> row (r + 8*(lane/16)), col = colTile*16 + nl
    const int mo = (lane >> 4) << 3;
    float* __restrict__ hrow = H + (size_t)(row0 + mo) * HID + nl;
    if (row0 + 16 <= n) {                         // tile fully in range: no per-store guards
        #pragma unroll
        for (int r = 0; r < 8; ++r) {
            float* __restrict__ o = hrow + (size_t)r * HID;
            o[ 0] = acc0[r];
            o[16] = acc1[r];
            o[32] = acc2[r];
            o[48] = acc3[r];
        }
    } else {                                      // ragged tail (unused when n % 16 == 0)
        #pragma unroll
        for (int r = 0; r < 8; ++r) {
            if (row0 + mo + r < n) {
                float* __restrict__ o = hrow + (size_t)r * HID;
                o[ 0] = acc0[r];
                o[16] = acc1[r];
                o[32] = acc2[r];
                o[48] = acc3[r];
            }
        }
    }
}

// ---------------------------------------------------------------------------
// Edge scatter: 16 threads per edge, each owns a float4 column chunk.
// agg[dst][c..c+3] += h[src][c..c+3] * dinv[src]*dinv[dst]
// Consecutive threads share an edge -> coalesced 256B gather per edge;
// atomics resolve in the 192MB L2 (all node data is L2-resident).
// ---------------------------------------------------------------------------
__global__ void __launch_bounds__(256)
gcn_scatter(const int* __restrict__ src, const int* __restrict__ dst,
            const float* __restrict__ dinv, const float* __restrict__ H,
            float* __restrict__ agg, int E) {
    long long t = (long long)blockIdx.x * blockDim.x + threadIdx.x;
    int e = (int)(t >> 4);
    if (e >= E) return;
    int c = ((int)t & 15) << 2;
    int s = src[e], d = dst[e];
    float norm = dinv[s] * dinv[d];
    const float4 hv = *(const float4*)(H + (size_t)s * HID + c);
    float* o = agg + (size_t)d * HID + c;
    atomicAdd(o + 0, hv.x * norm);
    atomicAdd(o + 1, hv.y * norm);
    atomicAdd(o + 2, hv.z * norm);
    atomicAdd(o + 3, hv.w * norm);
}

// ---------------------------------------------------------------------------
// Finalize: out = agg + h * dinv^2 (self-loop) + bias, optional ReLU
// ---------------------------------------------------------------------------
__global__ void __launch_bounds__(256)
gcn_finalize(const float* __restrict__ agg, const float* __restrict__ H,
             const float* __restrict__ dinv, const float* __restrict__ bias,
             float* __restrict__ out, int n, int relu) {
    long long t = (long long)blockIdx.x * blockDim.x + threadIdx.x;
    if (t >= (long long)n * HID) return;
    int node = (int)(t >> 6);
    int col  = (int)t & 63;
    float di = dinv[node];
    float v = agg[t] + H[t] * di * di + bias[col];
    if (relu) v = fmaxf(v, 0.0f);
    out[t] = v;
}

// ---------------------------------------------------------------------------
extern "C" void kernel_launch(void* const* d_in, const int* in_sizes, int n_in,
                              void* d_out, int out_size, void* d_ws, size_t ws_size,
                              hipStream_t stream) {
    const int*   edges = (const int*)d_in[0];
    const float* emb   = (const float*)d_in[1];
    const float* W1    = (const float*)d_in[2];
    const float* b1    = (const float*)d_in[3];
    const float* W2    = (const float*)d_in[4];
    const float* b2    = (const float*)d_in[5];

    const int E = in_sizes[0] / 2;
    const int N = in_sizes[1] / HID;
    const int* src = edges;
    const int* dst = edges + E;

    // Workspace layout (floats): dinv[N] | H[N*64] | agg[N*64] | X1[N*64]
    float* dinv = (float*)d_ws;
    float* H    = dinv + N;
    float* agg  = H + (size_t)N * HID;
    float* X1   = agg + (size_t)N * HID;

    const int THREADS = 256;
    const int gemmBlocks  = (N + 127) / 128;
    const int edgeBlocks  = (int)(((long long)E * 16 + THREADS - 1) / THREADS);
    const int nodeBlocks  = (int)(((long long)N * HID + THREADS - 1) / THREADS);

    // Degree / normalization (shared by both layers)
    hipMemsetAsync(dinv, 0, (size_t)N * sizeof(float), stream);
    gcn_deg_count<<<(E + THREADS - 1) / THREADS, THREADS, 0, stream>>>(dst, dinv, E);
    gcn_deg_finish<<<(N + THREADS - 1) / THREADS, THREADS, 0, stream>>>(dinv, N);

    // ---- Layer 1 ----
    gcn_gemm64_wmma<<<gemmBlocks, THREADS, 0, stream>>>(emb, W1, H, N);
    hipMemsetAsync(agg, 0, (size_t)N * HID * sizeof(float), stream);
    gcn_scatter<<<edgeBlocks, THREADS, 0, stream>>>(src, dst, dinv, H, agg, E);
    gcn_finalize<<<nodeBlocks, THREADS, 0, stream>>>(agg, H, dinv, b1, X1, N, 1);

    // ---- Layer 2 ----
    gcn_gemm64_wmma<<<gemmBlocks, THREADS, 0, stream>>>(X1, W2, H, N);
    hipMemsetAsync(agg, 0, (size_t)N * HID * sizeof(float), stream);
    gcn_scatter<<<edgeBlocks, THREADS, 0, stream>>>(src, dst, dinv, H, agg, E);
    gcn_finalize<<<nodeBlocks, THREADS, 0, stream>>>(agg, H, dinv, b2, (float*)d_out, N, 0);
}